// WeightDiscrepancyLoss_7456063226588
// MI455X (gfx1250) — compile-verified
//
#include <hip/hip_runtime.h>

typedef __attribute__((ext_vector_type(2))) float v2f;
typedef __attribute__((ext_vector_type(4))) float v4f;
typedef __attribute__((ext_vector_type(8))) float v8f;

// ---------------------------------------------------------------------------
// Wave32 reduction using the CDNA5 f32 WMMA (16x16x4) as a reducer.
// A-matrix layout (32-bit A, 16x4): lanes 0-15 hold rows M=0..15 with
// VGPR0 = K0, VGPR1 = K1; lanes 16-31 hold K2/K3 of the same rows.
// We place the per-lane partial in A[*][K0/K2] and zero in K1/K3, and use an
// all-ones B (layout-independent). Then D[i][j] = x_i + x_{i+16} for all j.
// Each lane sums its 8 D VGPRs (rows 0-7 in lanes 0-15, rows 8-15 in lanes
// 16-31), and one shfl_xor(16) produces the full 32-lane sum in every lane.
// EXEC must be all ones here: callers invoke this with no prior divergence.
// ---------------------------------------------------------------------------
__device__ __forceinline__ float wave_sum_wmma(float x) {
  v2f a;    a[0] = x;   a[1] = 0.0f;
  v2f ones; ones[0] = 1.0f; ones[1] = 1.0f;
  v8f c = {};
  // (neg_a, A, neg_b, B, c_mod, C, reuse_a, reuse_b)
  c = __builtin_amdgcn_wmma_f32_16x16x4_f32(false, a, false, ones,
                                            (short)0, c, false, false);
  float s = ((c[0] + c[1]) + (c[2] + c[3])) + ((c[4] + c[5]) + (c[6] + c[7]));
  s += __shfl_xor(s, 16, 32);
  return s;
}

// Streaming accumulation over one tensor pair with 128-bit non-temporal loads.
__device__ __forceinline__ void accum_pair(const float* __restrict__ pa,
                                           const float* __restrict__ pb,
                                           int n, int tid, int stride,
                                           float& dot, float& s1, float& s2) {
  const v4f* __restrict__ a = reinterpret_cast<const v4f*>(pa);
  const v4f* __restrict__ b = reinterpret_cast<const v4f*>(pb);
  const int n4 = n >> 2;
  for (int i = tid; i < n4; i += stride) {
    v4f x = __builtin_nontemporal_load(a + i);
    v4f y = __builtin_nontemporal_load(b + i);
    dot = fmaf(x[0], y[0], dot);
    dot = fmaf(x[1], y[1], dot);
    dot = fmaf(x[2], y[2], dot);
    dot = fmaf(x[3], y[3], dot);
    s1  = fmaf(x[0], x[0], s1);
    s1  = fmaf(x[1], x[1], s1);
    s1  = fmaf(x[2], x[2], s1);
    s1  = fmaf(x[3], x[3], s1);
    s2  = fmaf(y[0], y[0], s2);
    s2  = fmaf(y[1], y[1], s2);
    s2  = fmaf(y[2], y[2], s2);
    s2  = fmaf(y[3], y[3], s2);
  }
  // Scalar tail (sizes here are all multiples of 4, but stay general).
  const int rem = n & 3;
  if (rem && tid == 0) {
    for (int j = n - rem; j < n; ++j) {
      float xa = pa[j], xb = pb[j];
      dot = fmaf(xa, xb, dot);
      s1  = fmaf(xa, xa, s1);
      s2  = fmaf(xb, xb, s2);
    }
  }
}

__global__ void cosim_init_kernel(float* ws) {
  ws[0] = 0.0f; ws[1] = 0.0f; ws[2] = 0.0f;
}

__global__ __launch_bounds__(256) void cosim_main_kernel(
    const float* __restrict__ a0, const float* __restrict__ a1,
    const float* __restrict__ a2, const float* __restrict__ a3,
    const float* __restrict__ b0, const float* __restrict__ b1,
    const float* __restrict__ b2, const float* __restrict__ b3,
    int n0, int n1, int n2, int n3, float* __restrict__ ws) {
  const int tid    = blockIdx.x * blockDim.x + threadIdx.x;
  const int stride = gridDim.x * blockDim.x;

  float dot = 0.0f, s1 = 0.0f, s2 = 0.0f;
  accum_pair(a0, b0, n0, tid, stride, dot, s1, s2);
  accum_pair(a1, b1, n1, tid, stride, dot, s1, s2);
  accum_pair(a2, b2, n2, tid, stride, dot, s1, s2);
  accum_pair(a3, b3, n3, tid, stride, dot, s1, s2);

  // --- wave32 reduction via V_WMMA_F32_16X16X4_F32 (no divergence yet) ---
  float wd = wave_sum_wmma(dot);
  float w1 = wave_sum_wmma(s1);
  float w2 = wave_sum_wmma(s2);

  __shared__ float red[3][8];                // 256 threads -> 8 waves (wave32)
  const int wave = threadIdx.x >> 5;
  const int lane = threadIdx.x & 31;
  if (lane == 0) {
    red[0][wave] = wd;
    red[1][wave] = w1;
    red[2][wave] = w2;
  }
  __syncthreads();
  if (threadIdx.x == 0) {
    const int nwaves = blockDim.x >> 5;
    float td = 0.0f, t1 = 0.0f, t2 = 0.0f;
    for (int w = 0; w < nwaves; ++w) {
      td += red[0][w]; t1 += red[1][w]; t2 += red[2][w];
    }
    atomicAdd(&ws[0], td);
    atomicAdd(&ws[1], t1);
    atomicAdd(&ws[2], t2);
  }
}

__global__ void cosim_finalize_kernel(const float* __restrict__ ws,
                                      float* __restrict__ out) {
  out[0] = ws[0] / (sqrtf(ws[1]) * sqrtf(ws[2])) + 1.0f;
}

extern "C" void kernel_launch(void* const* d_in, const int* in_sizes, int n_in,
                              void* d_out, int out_size, void* d_ws, size_t ws_size,
                              hipStream_t stream) {
  (void)n_in; (void)out_size; (void)ws_size;
  const float* a0 = (const float*)d_in[0];
  const float* a1 = (const float*)d_in[1];
  const float* a2 = (const float*)d_in[2];
  const float* a3 = (const float*)d_in[3];
  const float* b0 = (const float*)d_in[4];
  const float* b1 = (const float*)d_in[5];
  const float* b2 = (const float*)d_in[6];
  const float* b3 = (const float*)d_in[7];
  float* ws  = (float*)d_ws;
  float* out = (float*)d_out;

  cosim_init_kernel<<<1, 1, 0, stream>>>(ws);

  // 4096 blocks x 256 threads = 1,048,576 threads: plenty of waves to saturate
  // 23.3 TB/s HBM; largest tensor (65.5M elems -> 16.4M float4) gives ~16
  // float4-pair iterations per thread.
  cosim_main_kernel<<<4096, 256, 0, stream>>>(a0, a1, a2, a3, b0, b1, b2, b3,
                                              in_sizes[0], in_sizes[1],
                                              in_sizes[2], in_sizes[3], ws);

  cosim_finalize_kernel<<<1, 1, 0, stream>>>(ws, out);
}